// PMG_7834020348697
// MI455X (gfx1250) — compile-verified
//
#include <hip/hip_runtime.h>
#include <hip/hip_bf16.h>
#include <math.h>

typedef __attribute__((ext_vector_type(2))) float v2f;
typedef __attribute__((ext_vector_type(8))) float v8f;

#define NANCH  1614
#define BATCH  16
#define TOPN   6
#define MCH    128
#define KBLK   144          /* 16 input channels x 9 taps per K-block */
#define NCOLS  64           /* output columns per workgroup (4 N-tiles of 16) */

/* ---------------- anchor generation (matches numpy _gen_anchors + PAD, int trunc) -------- */
__global__ void pmg_gen_anchors(float* __restrict__ boxes, float* __restrict__ areas)
{
    int i = blockIdx.x * blockDim.x + threadIdx.x;
    if (i >= NANCH) return;
    const double S1 = 1.2599210498948731906665444;   /* 2^(1/3) */
    const double S2 = 1.5874010519681993613971827;   /* 2^(2/3) */
    const double ARS[3] = {0.667, 1.0, 1.5};
    int stride, size, a, oy, ox;
    double sc;
    if (i < 1176)      { stride = 32;  size = 48;  int r = i;        a = r / 196; int p = r % 196; oy = p / 14; ox = p % 14; sc = (a < 3) ? S1 : S2; }
    else if (i < 1470) { stride = 64;  size = 96;  int r = i - 1176; a = r / 49;  int p = r % 49;  oy = p / 7;  ox = p % 7;  sc = (a < 3) ? S1 : S2; }
    else               { stride = 128; size = 192; int r = i - 1470; a = r / 16;  int p = r % 16;  oy = p / 4;  ox = p % 4;  sc = (a < 3) ? 1.0 : ((a < 6) ? S1 : S2); }
    double ar = ARS[a % 3];
    double hh = (double)size * sc / sqrt(ar);
    double ww = (double)size * sc * sqrt(ar);
    float cy = (float)stride * 0.5f + (float)(stride * oy);
    float cx = (float)stride * 0.5f + (float)(stride * ox);
    float e0 = (cy - (float)(hh * 0.5)) + 224.0f;
    float e1 = (cx - (float)(ww * 0.5)) + 224.0f;
    float e2 = (cy + (float)(hh * 0.5)) + 224.0f;
    float e3 = (cx + (float)(ww * 0.5)) + 224.0f;
    float b0 = (float)(int)e0, b1 = (float)(int)e1, b2 = (float)(int)e2, b3 = (float)(int)e3;
    boxes[i * 4 + 0] = b0; boxes[i * 4 + 1] = b1;
    boxes[i * 4 + 2] = b2; boxes[i * 4 + 3] = b3;
    areas[i] = (b2 - b0) * (b3 - b1);
}

/* ---------------- 3x3 conv (pad=1) + bias + relu, implicit GEMM on V_WMMA_F32_16X16X4_F32 ---
 * GEMM view: D[M=128][N=B*OH*OW] = W[M][K=Cin*9] x Im2col[K][N]
 * Workgroup tile: 128(M) x 64(N); wave w owns M-rows [16w,16w+16) and 4 N-tiles (4 acc chains).
 * LDS im2col panel is stored ROW-PAIR INTERLEAVED: element (k, col) lives at
 *   bpan[(k>>1)*2*NCOLS + col*2 + (k&1)]
 * so a WMMA B fragment {B[k][n], B[k+1][n]} (k even) is one aligned ds_load_b64.
 * A loads are software-pipelined one k-step ahead; unroll capped to stay <256 VGPRs. */
__global__ __launch_bounds__(256) void pmg_conv3x3_relu_wmma(
    const float* __restrict__ in, const float* __restrict__ wt,
    const float* __restrict__ bias, float* __restrict__ out,
    int Cin, int H, int W, int OH, int OW, int stride)
{
    __shared__ float bpan[KBLK * NCOLS];     /* 36 KB, pair-interleaved */

    const int tid   = threadIdx.x;
    const int lane  = tid & 31;
    const int wave  = tid >> 5;              /* 0..7  -> M tile */
    const int mbase = wave * 16;
    const int nbase = blockIdx.x * NCOLS;
    const int K     = Cin * 9;
    const int OHW   = OH * OW;
    const int Ntot  = BATCH * OHW;

    /* staging role: thread fills column scol, rows srow, srow+4, ... of the panel */
    const int scol = tid & 63;
    const int srow = tid >> 6;               /* 0..3 */
    int sn  = nbase + scol; if (sn > Ntot - 1) sn = Ntot - 1;   /* clamp (masked at store) */
    int sb  = sn / OHW;
    int sp  = sn % OHW;
    int oyS = (sp / OW) * stride;
    int oxS = (sp % OW) * stride;
    const float* inB = in + (size_t)sb * Cin * H * W;

    /* WMMA lane mapping (f32 16x16x4): A lane=M, pair=(lane>>4)*2 ; B lane=N, same pair */
    const int acol  = lane & 15;
    const int apair = (lane >> 4) * 2;
    const float* wrow = wt + (size_t)(mbase + acol) * K + apair;

    v8f acc0 = {0.f,0.f,0.f,0.f,0.f,0.f,0.f,0.f};
    v8f acc1 = acc0, acc2 = acc0, acc3 = acc0;

    const int nkb = Cin / 16;
    for (int kb = 0; kb < nkb; ++kb) {
        /* ---- stage im2col panel (KBLK rows x 64 cols, pair-interleaved) into LDS ---- */
        {
            int cibase = kb * 16;
            #pragma unroll
            for (int j = 0; j < KBLK / 4; ++j) {
                int r  = srow + j * 4;
                int ci = cibase + r / 9;
                int rr = r % 9;
                int iy = oyS + rr / 3 - 1;
                int ix = oxS + rr % 3 - 1;
                float v = 0.f;
                if ((unsigned)iy < (unsigned)H && (unsigned)ix < (unsigned)W)
                    v = inB[((size_t)ci * H + iy) * W + ix];
                bpan[(r >> 1) * (2 * NCOLS) + scol * 2 + (r & 1)] = v;
            }
        }
        __syncthreads();

        const float* wp = wrow + (size_t)kb * KBLK;
        __builtin_prefetch(wp + KBLK, 0, 1);          /* global_prefetch next weight block */
        float2 aNext = *(const float2*)(wp);          /* pipelined A fragment */
        #pragma unroll 6
        for (int k4 = 0; k4 < KBLK / 4; ++k4) {
            v2f a; a[0] = aNext.x; a[1] = aNext.y;
            if (k4 + 1 < KBLK / 4)
                aNext = *(const float2*)(wp + (k4 + 1) * 4);
            /* B fragments: one ds_load_b64 each (pair rows are adjacent in LDS) */
            const float2* bp = (const float2*)&bpan[(k4 * 2 + (apair >> 1)) * (2 * NCOLS)];
            float2 b0v = bp[acol +  0];
            float2 b1v = bp[acol + 16];
            float2 b2v = bp[acol + 32];
            float2 b3v = bp[acol + 48];
            v2f b0; b0[0] = b0v.x; b0[1] = b0v.y;
            v2f b1; b1[0] = b1v.x; b1[1] = b1v.y;
            v2f b2; b2[0] = b2v.x; b2[1] = b2v.y;
            v2f b3; b3[0] = b3v.x; b3[1] = b3v.y;
            acc0 = __builtin_amdgcn_wmma_f32_16x16x4_f32(false, a, false, b0, (short)0, acc0, false, false);
            acc1 = __builtin_amdgcn_wmma_f32_16x16x4_f32(false, a, false, b1, (short)0, acc1, false, false);
            acc2 = __builtin_amdgcn_wmma_f32_16x16x4_f32(false, a, false, b2, (short)0, acc2, false, false);
            acc3 = __builtin_amdgcn_wmma_f32_16x16x4_f32(false, a, false, b3, (short)0, acc3, false, false);
        }
        __syncthreads();
    }

    /* ---- C/D layout: VGPR i holds M=i (lanes 0-15) and M=i+8 (lanes 16-31), N=lane&15 ---- */
    const int mo = mbase + ((lane >> 4) << 3);
    #pragma unroll
    for (int t = 0; t < 4; ++t) {
        v8f acc = (t == 0) ? acc0 : (t == 1) ? acc1 : (t == 2) ? acc2 : acc3;
        int n = nbase + t * 16 + acol;
        if (n < Ntot) {
            int b = n / OHW, p = n % OHW;
            #pragma unroll
            for (int i = 0; i < 8; ++i) {
                int m = mo + i;
                float v = acc[i] + bias[m];
                v = fmaxf(v, 0.f);
                out[((size_t)(b * MCH + m)) * OHW + p] = v;
            }
        }
    }
}

/* ---------------- 1x1 "t" convs -> concatenated rpn scores (B x 1614) -------------------- */
__global__ void pmg_tconv(const float* __restrict__ d1, const float* __restrict__ d2,
                          const float* __restrict__ d3,
                          const float* __restrict__ w1, const float* __restrict__ bt1,
                          const float* __restrict__ w2, const float* __restrict__ bt2,
                          const float* __restrict__ w3, const float* __restrict__ bt3,
                          float* __restrict__ rpn)
{
    int g = blockIdx.x * blockDim.x + threadIdx.x;
    if (g >= BATCH * NANCH) return;
    int b = g / NANCH, r = g % NANCH;
    const float* src; const float* w; float bias; int p, np;
    if (r < 1176)      { int a = r / 196;          p = r % 196;          np = 196; src = d1 + (size_t)b * MCH * 196; w = w1 + a * MCH; bias = bt1[a]; }
    else if (r < 1470) { int a = (r - 1176) / 49;  p = (r - 1176) % 49;  np = 49;  src = d2 + (size_t)b * MCH * 49;  w = w2 + a * MCH; bias = bt2[a]; }
    else               { int a = (r - 1470) / 16;  p = (r - 1470) % 16;  np = 16;  src = d3 + (size_t)b * MCH * 16;  w = w3 + a * MCH; bias = bt3[a]; }
    float s = bias;
    #pragma unroll 8
    for (int c = 0; c < MCH; ++c)
        s += w[c] * src[(size_t)c * np + p];
    rpn[(size_t)b * NANCH + r] = s;
}

/* ---------------- hard NMS: one workgroup per image, 6 rounds of argmax + suppress ------- */
__global__ __launch_bounds__(256) void pmg_nms(const float* __restrict__ rpn,
                                               const float* __restrict__ boxes,
                                               const float* __restrict__ areas,
                                               float* __restrict__ top_prob,
                                               int* __restrict__ top_idx)
{
    __shared__ float sc[NANCH];
    __shared__ float rs[256];
    __shared__ int   ri[256];
    __shared__ int   sel;
    __shared__ float sb[5];      /* box(4), area */
    const int b = blockIdx.x, tid = threadIdx.x;
    const float* rb = rpn + (size_t)b * NANCH;

    for (int j = tid; j < NANCH; j += 256) sc[j] = rb[j];
    __syncthreads();

    for (int t = 0; t < TOPN; ++t) {
        /* argmax with first-index tie-break (matches jnp.argmax) */
        float best = -__builtin_inff(); int bi = 0x7fffffff;
        for (int j = tid; j < NANCH; j += 256) {
            float v = sc[j];
            if (v > best || (v == best && j < bi)) { best = v; bi = j; }
        }
        rs[tid] = best; ri[tid] = bi;
        __syncthreads();
        for (int s = 128; s > 0; s >>= 1) {
            if (tid < s) {
                float v = rs[tid + s]; int i2 = ri[tid + s];
                if (v > rs[tid] || (v == rs[tid] && i2 < ri[tid])) { rs[tid] = v; ri[tid] = i2; }
            }
            __syncthreads();
        }
        if (tid == 0) {
            int i = ri[0];
            sel = i;
            top_idx[b * TOPN + t]  = i;
            top_prob[b * TOPN + t] = rb[i];
            sb[0] = boxes[i * 4 + 0]; sb[1] = boxes[i * 4 + 1];
            sb[2] = boxes[i * 4 + 2]; sb[3] = boxes[i * 4 + 3];
            sb[4] = areas[i];
        }
        __syncthreads();
        float b0 = sb[0], b1 = sb[1], b2 = sb[2], b3 = sb[3], ai = sb[4];
        for (int j = tid; j < NANCH; j += 256) {
            float ih = fmaxf(fminf(boxes[j * 4 + 2], b2) - fmaxf(boxes[j * 4 + 0], b0), 0.f);
            float iw = fmaxf(fminf(boxes[j * 4 + 3], b3) - fmaxf(boxes[j * 4 + 1], b1), 0.f);
            float inter = ih * iw;
            float iou = inter / (areas[j] + ai - inter);
            if (iou >= 0.25f) sc[j] = -__builtin_inff();
        }
        __syncthreads();
        (void)sel;
    }
}

/* ---------------- bilinear crop-resize of padded image (padding handled implicitly) ------ */
__device__ __forceinline__ float pmg_fetch(const float* __restrict__ x, int b, int c, int y, int xi)
{
    int yy = y - 224, xx = xi - 224;
    if ((unsigned)yy < 448u && (unsigned)xx < 448u)
        return x[(((size_t)b * 3 + c) * 448 + yy) * 448 + xx];
    return 0.f;
}

__global__ __launch_bounds__(256) void pmg_crop(const float* __restrict__ x,
                                                const float* __restrict__ boxes,
                                                const int* __restrict__ top_idx,
                                                float* __restrict__ parts)
{
    const int blk   = blockIdx.x;
    const int bt    = blk / 196;                 /* 0..95 : (b*6 + t) */
    const int chunk = blk % 196;
    const int pix   = chunk * 256 + threadIdx.x; /* 0..50175 */
    const int py = pix / 224, px = pix % 224;
    const int b  = bt / TOPN;
    const int idx = top_idx[bt];
    float b0 = boxes[idx * 4 + 0], b1 = boxes[idx * 4 + 1];
    float b2 = boxes[idx * 4 + 2], b3 = boxes[idx * 4 + 3];
    float ty = (float)py * (1.0f / 223.0f);
    float tx = (float)px * (1.0f / 223.0f);
    float ys = b0 + ty * (b2 - b0 - 1.0f);
    float xs = b1 + tx * (b3 - b1 - 1.0f);
    int y0 = min(max((int)floorf(ys), 0), 895);
    int x0 = min(max((int)floorf(xs), 0), 895);
    int y1 = min(y0 + 1, 895);
    int x1 = min(x0 + 1, 895);
    float wy = ys - (float)y0;
    float wx = xs - (float)x0;
    #pragma unroll
    for (int c = 0; c < 3; ++c) {
        float v00 = pmg_fetch(x, b, c, y0, x0);
        float v01 = pmg_fetch(x, b, c, y0, x1);
        float v10 = pmg_fetch(x, b, c, y1, x0);
        float v11 = pmg_fetch(x, b, c, y1, x1);
        float v = (1.f - wy) * ((1.f - wx) * v00 + wx * v01)
                +        wy  * ((1.f - wx) * v10 + wx * v11);
        parts[(((size_t)bt * 3 + c) * 224 + py) * 224 + px] = v;
    }
}

/* ---------------- driver ----------------------------------------------------------------- */
extern "C" void kernel_launch(void* const* d_in, const int* in_sizes, int n_in,
                              void* d_out, int out_size, void* d_ws, size_t ws_size,
                              hipStream_t stream)
{
    (void)in_sizes; (void)n_in; (void)out_size; (void)ws_size;
    const float* x   = (const float*)d_in[0];
    const float* f3  = (const float*)d_in[1];
    const float* wd1 = (const float*)d_in[2];
    const float* bd1 = (const float*)d_in[3];
    const float* wd2 = (const float*)d_in[4];
    const float* bd2 = (const float*)d_in[5];
    const float* wd3 = (const float*)d_in[6];
    const float* bd3 = (const float*)d_in[7];
    const float* wt1 = (const float*)d_in[8];
    const float* bt1 = (const float*)d_in[9];
    const float* wt2 = (const float*)d_in[10];
    const float* bt2 = (const float*)d_in[11];
    const float* wt3 = (const float*)d_in[12];
    const float* bt3 = (const float*)d_in[13];

    float* ws    = (float*)d_ws;
    float* d1    = ws;               /* 16*128*196 = 401408 */
    float* d2    = d1 + 401408;      /* 16*128*49  = 100352 */
    float* d3    = d2 + 100352;      /* 16*128*16  =  32768 */
    float* rpn   = d3 + 32768;       /* 16*1614    =  25824 */
    float* boxes = rpn + 25824;      /* 1614*4     =   6456 */
    float* areas = boxes + 6456;     /* 1614 */
    int*   tidx  = (int*)(areas + NANCH);   /* 96 ints */

    float* top_prob = (float*)d_out;        /* 96 floats */
    float* parts    = top_prob + BATCH * TOPN;

    pmg_gen_anchors<<<(NANCH + 255) / 256, 256, 0, stream>>>(boxes, areas);

    /* d1: 2048->128, 14x14, stride 1 : N = 3136 -> 49 wgs of 64 cols */
    pmg_conv3x3_relu_wmma<<<49, 256, 0, stream>>>(f3, wd1, bd1, d1, 2048, 14, 14, 14, 14, 1);
    /* d2: 128->128, 14x14 -> 7x7, stride 2 : N = 784 -> 13 wgs (last masked) */
    pmg_conv3x3_relu_wmma<<<13, 256, 0, stream>>>(d1, wd2, bd2, d2, 128, 14, 14, 7, 7, 2);
    /* d3: 128->128, 7x7 -> 4x4, stride 2 : N = 256 -> 4 wgs */
    pmg_conv3x3_relu_wmma<<<4, 256, 0, stream>>>(d2, wd3, bd3, d3, 128, 7, 7, 4, 4, 2);

    pmg_tconv<<<(BATCH * NANCH + 255) / 256, 256, 0, stream>>>(
        d1, d2, d3, wt1, bt1, wt2, bt2, wt3, bt3, rpn);

    pmg_nms<<<BATCH, 256, 0, stream>>>(rpn, boxes, areas, top_prob, tidx);

    pmg_crop<<<BATCH * TOPN * 196, 256, 0, stream>>>(x, boxes, tidx, parts);
}